// GAT_25237227832044
// MI455X (gfx1250) — compile-verified
//
#include <hip/hip_runtime.h>
#include <hip/hip_bf16.h>

#define N_NODES   100000
#define N_EDGES   3200000
#define E_TOT     (N_EDGES + N_NODES)   // with self loops
#define IN_CH     128
#define HID       8
#define OUT_CH    2
#define NUM_GRAPHS 64
#define NEG_SLOPE 0.2f

typedef float v2f __attribute__((ext_vector_type(2)));
typedef float v8f __attribute__((ext_vector_type(8)));

// ---- monotone float<->uint encoding for atomic max on floats ----
__device__ __forceinline__ unsigned enc_f(float f) {
  unsigned u = __float_as_uint(f);
  return (u & 0x80000000u) ? ~u : (u | 0x80000000u);
}
__device__ __forceinline__ float dec_f(unsigned u) {
  return (u & 0x80000000u) ? __uint_as_float(u & 0x7FFFFFFFu) : __uint_as_float(~u);
}
#define ENC_NEG_INF 0x007FFFFFu   // enc(-inf)

__device__ __forceinline__ float leaky(float z) { return z > 0.f ? z : NEG_SLOPE * z; }

// ---------------- init scratch ----------------
__global__ void gat_init(float* __restrict__ agg1, float* __restrict__ s1,
                         unsigned* __restrict__ m1, float* __restrict__ s2,
                         unsigned* __restrict__ m2, float* __restrict__ agg2,
                         float* __restrict__ pool, float* __restrict__ cnt) {
  int i = blockIdx.x * blockDim.x + threadIdx.x;
  if (i < N_NODES * 64) agg1[i] = 0.f;
  if (i < N_NODES * 8)  { s1[i] = 0.f; m1[i] = ENC_NEG_INF; }
  if (i < N_NODES * 2)  agg2[i] = 0.f;
  if (i < N_NODES)      { s2[i] = 0.f; m2[i] = ENC_NEG_INF; }
  if (i < NUM_GRAPHS * OUT_CH) pool[i] = 0.f;
  if (i < NUM_GRAPHS)   cnt[i] = 0.f;
}

// ---------------- h1 = x @ W1 via V_WMMA_F32_16X16X4_F32 ----------------
// One wave computes a 16x64 output tile (4 accumulators of 16x16), 32 K-steps.
// W1 is staged in LDS K-pair-packed so each B operand is a single ds_load_b64:
//   sW[(k/2)*128 + 2*n + (k&1)] = W1[k*64 + n]
__global__ __launch_bounds__(256) void gat_gemm1(const float* __restrict__ x,
                                                 const float* __restrict__ W1,
                                                 float* __restrict__ h1) {
  __shared__ float sW[IN_CH * 64];            // 32 KB of the 320 KB WGP LDS
  for (int i = threadIdx.x; i < IN_CH * 64; i += 256) {
    int k = i >> 6, n = i & 63;               // coalesced read of W1
    sW[(k >> 1) * 128 + 2 * n + (k & 1)] = W1[i];
  }
  __syncthreads();

  const int lane = threadIdx.x & 31;
  const int wave = threadIdx.x >> 5;
  const int tile = blockIdx.x * 8 + wave;      // 16-row tile index
  if (tile * 16 >= N_NODES) return;            // wave-uniform: EXEC stays all-1s

  const int R    = tile * 16;
  const int m    = lane & 15;                  // M (A) / N (B,C,D) within tile
  const int g    = lane >> 4;                  // lane-group 0/1

  v8f acc0 = {}, acc1 = {}, acc2 = {}, acc3 = {};
  const float* xrow = x + (size_t)(R + m) * IN_CH;

  #pragma unroll 4
  for (int k0 = 0; k0 < IN_CH; k0 += 4) {
    const int kk = k0 + 2 * g;                 // this lane's K pair (even)
    v2f a = *(const v2f*)(xrow + kk);          // one global_load_b64
    const float* wp = sW + (kk >> 1) * 128;    // packed K-pair row
    v2f b0 = *(const v2f*)(wp + 2 * (m +  0)); // one ds_load_b64 each
    v2f b1 = *(const v2f*)(wp + 2 * (m + 16));
    v2f b2 = *(const v2f*)(wp + 2 * (m + 32));
    v2f b3 = *(const v2f*)(wp + 2 * (m + 48));
    acc0 = __builtin_amdgcn_wmma_f32_16x16x4_f32(false, a, false, b0, (short)0, acc0, false, false);
    acc1 = __builtin_amdgcn_wmma_f32_16x16x4_f32(false, a, false, b1, (short)0, acc1, false, false);
    acc2 = __builtin_amdgcn_wmma_f32_16x16x4_f32(false, a, false, b2, (short)0, acc2, false, false);
    acc3 = __builtin_amdgcn_wmma_f32_16x16x4_f32(false, a, false, b3, (short)0, acc3, false, false);
  }

  // C/D layout: VGPR v, lane-group g -> row = v + 8*g, col = 16*t + m
  #pragma unroll
  for (int v = 0; v < 8; ++v) {
    float* out = h1 + (size_t)(R + v + 8 * g) * 64;
    out[m +  0] = acc0[v];
    out[m + 16] = acc1[v];
    out[m + 32] = acc2[v];
    out[m + 48] = acc3[v];
  }
}

// ---------------- per-node attention coefficients, layer 1 ----------------
__global__ void gat_alpha1(const float* __restrict__ h1, const float* __restrict__ a_src,
                           const float* __restrict__ a_dst,
                           float* __restrict__ al_s, float* __restrict__ al_d) {
  int i = blockIdx.x * blockDim.x + threadIdx.x;    // (node, head)
  if (i >= N_NODES * 8) return;
  int n = i >> 3, h = i & 7;
  const float4* hp = (const float4*)(h1 + (size_t)n * 64 + h * 8);
  float4 h0 = hp[0], h1v = hp[1];
  const float4* as = (const float4*)(a_src + h * 8);
  const float4* ad = (const float4*)(a_dst + h * 8);
  float4 s0 = as[0], s1 = as[1], d0 = ad[0], d1 = ad[1];
  al_s[i] = h0.x*s0.x + h0.y*s0.y + h0.z*s0.z + h0.w*s0.w
          + h1v.x*s1.x + h1v.y*s1.y + h1v.z*s1.z + h1v.w*s1.w;
  al_d[i] = h0.x*d0.x + h0.y*d0.y + h0.z*d0.z + h0.w*d0.w
          + h1v.x*d1.x + h1v.y*d1.y + h1v.z*d1.z + h1v.w*d1.w;
}

__device__ __forceinline__ void edge_ends(const int* __restrict__ ei, int e, int& s, int& d) {
  if (e < N_EDGES) { s = ei[e]; d = ei[N_EDGES + e]; }
  else             { s = d = e - N_EDGES; }           // self loop
}

__device__ __forceinline__ void logits8(const float* al_s, const float* al_d,
                                        int s, int d, float* z) {
  const float4* ps = (const float4*)(al_s + s * 8);
  const float4* pd = (const float4*)(al_d + d * 8);
  float4 a0 = ps[0], a1 = ps[1], b0 = pd[0], b1 = pd[1];
  z[0] = leaky(a0.x + b0.x); z[1] = leaky(a0.y + b0.y);
  z[2] = leaky(a0.z + b0.z); z[3] = leaky(a0.w + b0.w);
  z[4] = leaky(a1.x + b1.x); z[5] = leaky(a1.y + b1.y);
  z[6] = leaky(a1.z + b1.z); z[7] = leaky(a1.w + b1.w);
}

// ---------------- layer-1 edge passes ----------------
__global__ void gat_edge_max1(const int* __restrict__ ei, const float* __restrict__ al_s,
                              const float* __restrict__ al_d, unsigned* __restrict__ m1) {
  int e = blockIdx.x * blockDim.x + threadIdx.x;
  if (e >= E_TOT) return;
  int s, d; edge_ends(ei, e, s, d);
  float z[8]; logits8(al_s, al_d, s, d, z);
  #pragma unroll
  for (int h = 0; h < 8; ++h) atomicMax(&m1[d * 8 + h], enc_f(z[h]));
}

__global__ void gat_edge_sum1(const int* __restrict__ ei, const float* __restrict__ al_s,
                              const float* __restrict__ al_d, const unsigned* __restrict__ m1,
                              float* __restrict__ s1) {
  int e = blockIdx.x * blockDim.x + threadIdx.x;
  if (e >= E_TOT) return;
  int s, d; edge_ends(ei, e, s, d);
  float z[8]; logits8(al_s, al_d, s, d, z);
  #pragma unroll
  for (int h = 0; h < 8; ++h)
    atomicAdd(&s1[d * 8 + h], expf(z[h] - dec_f(m1[d * 8 + h])));
}

__global__ void gat_edge_agg1(const int* __restrict__ ei, const float* __restrict__ al_s,
                              const float* __restrict__ al_d, const unsigned* __restrict__ m1,
                              const float* __restrict__ s1, const float* __restrict__ h1,
                              float* __restrict__ agg1) {
  int e = blockIdx.x * blockDim.x + threadIdx.x;
  if (e >= E_TOT) return;
  int s, d; edge_ends(ei, e, s, d);
  float z[8]; logits8(al_s, al_d, s, d, z);
  const float4* hs = (const float4*)(h1 + (size_t)s * 64);
  #pragma unroll
  for (int h = 0; h < 8; ++h) {
    float a = expf(z[h] - dec_f(m1[d * 8 + h])) / (s1[d * 8 + h] + 1e-16f);
    float4 v0 = hs[2 * h], v1 = hs[2 * h + 1];
    float* o = agg1 + (size_t)d * 64 + h * 8;
    atomicAdd(o + 0, v0.x * a); atomicAdd(o + 1, v0.y * a);
    atomicAdd(o + 2, v0.z * a); atomicAdd(o + 3, v0.w * a);
    atomicAdd(o + 4, v1.x * a); atomicAdd(o + 5, v1.y * a);
    atomicAdd(o + 6, v1.z * a); atomicAdd(o + 7, v1.w * a);
  }
}

// ---------------- bias + ELU (writes activated features back over h1) ----------------
__global__ void gat_bias_elu1(const float* __restrict__ agg1, const float* __restrict__ b1,
                              float* __restrict__ hact) {
  int i = blockIdx.x * blockDim.x + threadIdx.x;
  if (i >= N_NODES * 64) return;
  float v = agg1[i] + b1[i & 63];
  hact[i] = v > 0.f ? v : (expf(v) - 1.f);
}

// ---------------- layer 2: h2 = hact @ W2 (64x2) + attention coeffs ----------------
__global__ void gat_gemm2(const float* __restrict__ hact, const float* __restrict__ W2,
                          const float* __restrict__ a_src2, const float* __restrict__ a_dst2,
                          float* __restrict__ h2, float* __restrict__ al_s2,
                          float* __restrict__ al_d2) {
  int n = blockIdx.x * blockDim.x + threadIdx.x;
  if (n >= N_NODES) return;
  const float* r = hact + (size_t)n * 64;
  float a0 = 0.f, a1 = 0.f;
  #pragma unroll
  for (int j = 0; j < 64; ++j) { float v = r[j]; a0 += v * W2[j * 2]; a1 += v * W2[j * 2 + 1]; }
  h2[n * 2] = a0; h2[n * 2 + 1] = a1;
  al_s2[n] = a0 * a_src2[0] + a1 * a_src2[1];
  al_d2[n] = a0 * a_dst2[0] + a1 * a_dst2[1];
}

// ---------------- layer-2 edge passes (1 head, 2 channels) ----------------
__global__ void gat_edge_max2(const int* __restrict__ ei, const float* __restrict__ al_s,
                              const float* __restrict__ al_d, unsigned* __restrict__ m2) {
  int e = blockIdx.x * blockDim.x + threadIdx.x;
  if (e >= E_TOT) return;
  int s, d; edge_ends(ei, e, s, d);
  atomicMax(&m2[d], enc_f(leaky(al_s[s] + al_d[d])));
}

__global__ void gat_edge_sum2(const int* __restrict__ ei, const float* __restrict__ al_s,
                              const float* __restrict__ al_d, const unsigned* __restrict__ m2,
                              float* __restrict__ s2) {
  int e = blockIdx.x * blockDim.x + threadIdx.x;
  if (e >= E_TOT) return;
  int s, d; edge_ends(ei, e, s, d);
  atomicAdd(&s2[d], expf(leaky(al_s[s] + al_d[d]) - dec_f(m2[d])));
}

__global__ void gat_edge_agg2(const int* __restrict__ ei, const float* __restrict__ al_s,
                              const float* __restrict__ al_d, const unsigned* __restrict__ m2,
                              const float* __restrict__ s2, const float* __restrict__ h2,
                              float* __restrict__ agg2) {
  int e = blockIdx.x * blockDim.x + threadIdx.x;
  if (e >= E_TOT) return;
  int s, d; edge_ends(ei, e, s, d);
  float a = expf(leaky(al_s[s] + al_d[d]) - dec_f(m2[d])) / (s2[d] + 1e-16f);
  atomicAdd(&agg2[d * 2 + 0], h2[s * 2 + 0] * a);
  atomicAdd(&agg2[d * 2 + 1], h2[s * 2 + 1] * a);
}

// ---------------- global mean pool ----------------
__global__ void gat_pool(const int* __restrict__ batch, const float* __restrict__ agg2,
                         const float* __restrict__ b2, float* __restrict__ pool,
                         float* __restrict__ cnt) {
  int n = blockIdx.x * blockDim.x + threadIdx.x;
  if (n >= N_NODES) return;
  int g = batch[n];
  atomicAdd(&pool[g * 2 + 0], agg2[n * 2 + 0] + b2[0]);
  atomicAdd(&pool[g * 2 + 1], agg2[n * 2 + 1] + b2[1]);
  atomicAdd(&cnt[g], 1.f);
}

__global__ void gat_final(const float* __restrict__ pool, const float* __restrict__ cnt,
                          float* __restrict__ out) {
  int t = threadIdx.x;
  if (t < NUM_GRAPHS * OUT_CH) out[t] = pool[t] / fmaxf(cnt[t >> 1], 1.f);
}

extern "C" void kernel_launch(void* const* d_in, const int* in_sizes, int n_in,
                              void* d_out, int out_size, void* d_ws, size_t ws_size,
                              hipStream_t stream) {
  const float* x      = (const float*)d_in[0];
  const int*   ei     = (const int*)  d_in[1];
  const int*   batch  = (const int*)  d_in[2];
  const float* W1     = (const float*)d_in[3];
  const float* a_src1 = (const float*)d_in[4];
  const float* a_dst1 = (const float*)d_in[5];
  const float* b1     = (const float*)d_in[6];
  const float* W2     = (const float*)d_in[7];
  const float* a_src2 = (const float*)d_in[8];
  const float* a_dst2 = (const float*)d_in[9];
  const float* b2     = (const float*)d_in[10];
  float* out = (float*)d_out;

  // workspace carve (floats); all sub-buffers stay 16B aligned
  float*    h1    = (float*)d_ws;                 // N*64   (reused as ELU output)
  float*    agg1  = h1    + (size_t)N_NODES * 64; // N*64
  float*    al_s1 = agg1  + (size_t)N_NODES * 64; // N*8
  float*    al_d1 = al_s1 + (size_t)N_NODES * 8;  // N*8
  float*    s1    = al_d1 + (size_t)N_NODES * 8;  // N*8
  unsigned* m1    = (unsigned*)(s1 + (size_t)N_NODES * 8);   // N*8
  float*    h2    = (float*)(m1 + (size_t)N_NODES * 8);      // N*2
  float*    al_s2 = h2    + (size_t)N_NODES * 2;  // N
  float*    al_d2 = al_s2 + N_NODES;              // N
  float*    s2    = al_d2 + N_NODES;              // N
  unsigned* m2    = (unsigned*)(s2 + N_NODES);    // N
  float*    agg2  = (float*)(m2 + N_NODES);       // N*2
  float*    pool  = agg2  + (size_t)N_NODES * 2;  // 128
  float*    cnt   = pool  + NUM_GRAPHS * OUT_CH;  // 64

  const int TB = 256;
  const int gInit  = (N_NODES * 64 + TB - 1) / TB;
  const int gGemm1 = (N_NODES / 16 + 7) / 8;      // 8 waves (16-row tiles) per block
  const int gAl1   = (N_NODES * 8 + TB - 1) / TB;
  const int gEdge  = (E_TOT + TB - 1) / TB;
  const int gNode  = (N_NODES + TB - 1) / TB;

  gat_init<<<gInit, TB, 0, stream>>>(agg1, s1, m1, s2, m2, agg2, pool, cnt);
  gat_gemm1<<<gGemm1, TB, 0, stream>>>(x, W1, h1);
  gat_alpha1<<<gAl1, TB, 0, stream>>>(h1, a_src1, a_dst1, al_s1, al_d1);
  gat_edge_max1<<<gEdge, TB, 0, stream>>>(ei, al_s1, al_d1, m1);
  gat_edge_sum1<<<gEdge, TB, 0, stream>>>(ei, al_s1, al_d1, m1, s1);
  gat_edge_agg1<<<gEdge, TB, 0, stream>>>(ei, al_s1, al_d1, m1, s1, h1, agg1);
  gat_bias_elu1<<<gInit, TB, 0, stream>>>(agg1, b1, h1);   // h1 now holds elu(conv1)
  gat_gemm2<<<gNode, TB, 0, stream>>>(h1, W2, a_src2, a_dst2, h2, al_s2, al_d2);
  gat_edge_max2<<<gEdge, TB, 0, stream>>>(ei, al_s2, al_d2, m2);
  gat_edge_sum2<<<gEdge, TB, 0, stream>>>(ei, al_s2, al_d2, m2, s2);
  gat_edge_agg2<<<gEdge, TB, 0, stream>>>(ei, al_s2, al_d2, m2, s2, h2, agg2);
  gat_pool<<<gNode, TB, 0, stream>>>(batch, agg2, b2, pool, cnt);
  gat_final<<<1, 128, 0, stream>>>(pool, cnt, out);
}